// FCOSLoss_66468913873275
// MI455X (gfx1250) — compile-verified
//
#include <hip/hip_runtime.h>

#define F_EPS   1e-8f
#define F_ALPHA 0.25f

typedef __attribute__((ext_vector_type(2))) float v2f;
typedef __attribute__((ext_vector_type(8))) float v8f;

// ---------------------------------------------------------------------------
// Wave-level sum using V_WMMA_F32_16X16X4_F32 (CDNA5).
// A (16x4 f32): lanes 0-15 supply K=0 (=v) and K=1 (=0) for row M=lane,
//               lanes 16-31 supply K=2 (=v) and K=3 (=0) for row M=lane-16.
// B (4x16) = all ones  ->  D[i][j] = rowsum_i = v_i + v_{i+16}.
// D layout: lane L, VGPR r holds M = r + 8*(L>=16). Summing the 8 D VGPRs per
// lane gives sum(rowsums 0-7) on lanes 0-15 and sum(rowsums 8-15) on lanes
// 16-31; one shfl_xor(16) yields the full 32-lane sum on every lane.
// Requires EXEC all-1s: called from convergent control flow only.
// ---------------------------------------------------------------------------
__device__ __forceinline__ float wave_sum_wmma(float v) {
  v2f a; a[0] = v;    a[1] = 0.0f;
  v2f b; b[0] = 1.0f; b[1] = 1.0f;
  v8f c = {};
  c = __builtin_amdgcn_wmma_f32_16x16x4_f32(
      /*neg_a=*/false, a, /*neg_b=*/false, b,
      /*c_mod=*/(short)0, c, /*reuse_a=*/false, /*reuse_b=*/false);
  float s = c[0] + c[1] + c[2] + c[3] + c[4] + c[5] + c[6] + c[7];
  s += __shfl_xor(s, 16, 32);
  return s;
}

__device__ __forceinline__ float sigmoid_clamped(float x) {
  float p = 1.0f / (1.0f + __expf(-x));
  return fminf(fmaxf(p, F_EPS), 1.0f - F_EPS);
}

// focal term for a negative-class element: -(1-a) * p^2 * log(1-p)
__device__ __forceinline__ float focal_neg(float x) {
  float p = sigmoid_clamped(x);
  return -(1.0f - F_ALPHA) * p * p * __logf(1.0f - p);
}

// focal term for the positive (one-hot) element: -a * (1-p)^2 * log(p)
__device__ __forceinline__ float focal_pos(float x) {
  float p = sigmoid_clamped(x);
  float q = 1.0f - p;
  return -F_ALPHA * q * q * __logf(p);
}

// ---------------------------------------------------------------------------
// Pass 1: one thread per (image, location). Streams 20 float4 of cls logits,
// computes focal (neg branch for all classes + one-hot fixup), BCE centerness
// and GIoU on positives. Per-wave WMMA reduction, per-block LDS combine,
// deterministic per-block partial written to d_ws: partials[q][b][blockIdx.x].
// ---------------------------------------------------------------------------
__global__ void __launch_bounds__(256)
fcos_partial_kernel(const float* __restrict__ cls_logits,
                    const float* __restrict__ reg_preds,
                    const float* __restrict__ ctr_logits,
                    const int*   __restrict__ cls_targets,
                    const float* __restrict__ reg_targets,
                    const float* __restrict__ ctr_targets,
                    float* __restrict__ partials,
                    int N, int C, int B) {
  const int b    = blockIdx.y;
  const int n    = blockIdx.x * blockDim.x + threadIdx.x;
  const int nblk = gridDim.x;

  float cls_acc = 0.0f, ctr_acc = 0.0f, reg_acc = 0.0f, pos_acc = 0.0f;

  if (n < N) {
    const size_t loc = (size_t)b * (size_t)N + (size_t)n;

    // ---- classification focal loss over C classes (vectorized) ----
    const float4* cp = (const float4*)(cls_logits + loc * (size_t)C);
    const int C4 = C >> 2;  // C = 80 -> 20 float4 per location, 16B aligned
    float s = 0.0f;
#pragma unroll 5
    for (int i = 0; i < C4; ++i) {
      float4 x = cp[i];
      s += focal_neg(x.x) + focal_neg(x.y) + focal_neg(x.z) + focal_neg(x.w);
    }
    const int label = cls_targets[loc];       // 0..C ; one-hot hits class label-1
    if (label >= 1 && label <= C) {
      float xl = cls_logits[loc * (size_t)C + (size_t)(label - 1)];
      s += focal_pos(xl) - focal_neg(xl);     // swap neg-branch for pos-branch
    }
    cls_acc = s;

    // ---- centerness BCE + GIoU regression on positives only ----
    const float ct = ctr_targets[loc];
    if (ct > -1.0f) {
      pos_acc = 1.0f;

      float p = sigmoid_clamped(ctr_logits[loc]);
      ctr_acc = -(ct * __logf(p) + (1.0f - ct) * __logf(1.0f - p));

      float4 pr = *(const float4*)(reg_preds   + loc * 4);
      float4 tg = *(const float4*)(reg_targets + loc * 4);
      // (l,t,r,b) offsets
      float wmin = fmaxf(fminf(pr.x, tg.x) + fminf(pr.z, tg.z), 0.0f);
      float hmin = fmaxf(fminf(pr.y, tg.y) + fminf(pr.w, tg.w), 0.0f);
      float overlap = wmin * hmin;
      float area_p  = (pr.x + pr.z) * (pr.y + pr.w);
      float area_t  = (tg.x + tg.z) * (tg.y + tg.w);
      float uni     = fmaxf(area_p + area_t - overlap, F_EPS);
      float iou     = overlap / uni;
      float wmax = fmaxf(fmaxf(pr.x, tg.x) + fmaxf(pr.z, tg.z), 0.0f);
      float hmax = fmaxf(fmaxf(pr.y, tg.y) + fmaxf(pr.w, tg.w), 0.0f);
      float c_area = wmax * hmax;
      float giou = iou - (c_area - uni) / fmaxf(c_area, F_EPS);
      reg_acc = 1.0f - giou;
    }
  }

  // ---- per-wave WMMA reductions (convergent: every thread executes) ----
  float wcls = wave_sum_wmma(cls_acc);
  float wctr = wave_sum_wmma(ctr_acc);
  float wreg = wave_sum_wmma(reg_acc);
  float wpos = wave_sum_wmma(pos_acc);

  __shared__ float lds[4][8];            // 4 quantities x up to 8 waves
  const int wave = threadIdx.x >> 5;
  const int lane = threadIdx.x & 31;
  if (lane == 0) {
    lds[0][wave] = wcls;
    lds[1][wave] = wctr;
    lds[2][wave] = wreg;
    lds[3][wave] = wpos;
  }
  __syncthreads();

  const int nwaves = blockDim.x >> 5;
  if (threadIdx.x < 4) {
    float t = 0.0f;
    for (int w = 0; w < nwaves; ++w) t += lds[threadIdx.x][w];
    partials[((size_t)threadIdx.x * B + b) * nblk + blockIdx.x] = t;
  }
}

// ---------------------------------------------------------------------------
// Pass 2: deterministic fixed-order combine. 64 threads: thread t sums
// partials for quantity q=t/16, image b=t%16; thread 0 then applies the
// per-image num_pos normalization and batch means, writing
// [total, cls, reg, ctr] to d_out.
// ---------------------------------------------------------------------------
__global__ void fcos_finalize_kernel(const float* __restrict__ partials,
                                     float* __restrict__ out,
                                     int B, int nblk) {
  __shared__ float sums[4][16];
  const int t = threadIdx.x;
  if (t < 4 * B) {
    const int q = t / 16;
    const int b = t % 16;
    const float* p = partials + ((size_t)q * B + b) * nblk;
    float s = 0.0f;
    for (int i = 0; i < nblk; ++i) s += p[i];
    sums[q][b] = s;
  }
  __syncthreads();
  if (t == 0) {
    float cls_m = 0.0f, ctr_m = 0.0f, reg_m = 0.0f;
    for (int b = 0; b < B; ++b) {
      float inv_np = 1.0f / fmaxf(sums[3][b], 1.0f);
      cls_m += sums[0][b] * inv_np;
      ctr_m += sums[1][b] * inv_np;
      reg_m += sums[2][b] * inv_np;
    }
    const float invB = 1.0f / (float)B;
    cls_m *= invB; ctr_m *= invB; reg_m *= invB;
    out[0] = cls_m + reg_m + ctr_m;
    out[1] = cls_m;
    out[2] = reg_m;
    out[3] = ctr_m;
  }
}

extern "C" void kernel_launch(void* const* d_in, const int* in_sizes, int n_in,
                              void* d_out, int out_size, void* d_ws, size_t ws_size,
                              hipStream_t stream) {
  const float* cls_logits  = (const float*)d_in[0];
  const float* reg_preds   = (const float*)d_in[1];
  const float* ctr_logits  = (const float*)d_in[2];
  const int*   cls_targets = (const int*)  d_in[3];
  const float* reg_targets = (const float*)d_in[4];
  const float* ctr_targets = (const float*)d_in[5];
  float* out = (float*)d_out;

  const int B     = 16;                    // per reference setup
  const int total = in_sizes[3];           // B * N  (cls_targets flat count)
  const int N     = total / B;             // 17064
  const int C     = in_sizes[0] / total;   // 80

  const int TPB  = 256;                    // 8 wave32s per block
  const int nblk = (N + TPB - 1) / TPB;    // 67 blocks per image

  float* partials = (float*)d_ws;          // 4 * B * nblk floats (~17 KB)

  dim3 grid(nblk, B, 1);
  fcos_partial_kernel<<<grid, TPB, 0, stream>>>(
      cls_logits, reg_preds, ctr_logits, cls_targets, reg_targets, ctr_targets,
      partials, N, C, B);

  fcos_finalize_kernel<<<1, 64, 0, stream>>>(partials, out, B, nblk);
}